// LinearAttention_72103910965318
// MI455X (gfx1250) — compile-verified
//
#include <hip/hip_runtime.h>
#include <math.h>
#include <stdint.h>

// ---------------- problem constants ----------------
#define B_     8
#define DIM    256
#define HEADS  8
#define DH     64
#define HID    512          // HEADS*DH
#define NTOK   4096         // 64*64
#define ROWS   (B_ * NTOK)  // 32768
#define QKVC   (3 * HID)    // 1536
#define EPS_GN 1e-5f

// use global_load_async_to_lds for the qkv A-tile (set 0 to fall back)
#define QKV_ASYNC 1

typedef __attribute__((ext_vector_type(2))) float v2f;
typedef __attribute__((ext_vector_type(8))) float v8f;

static __device__ __forceinline__ v8f vzero8() {
    v8f z;
#pragma unroll
    for (int i = 0; i < 8; ++i) z[i] = 0.0f;
    return z;
}

// D = A(16x4 f32) * B(4x16 f32) + C(16x16 f32), wave32, full fp32 path
static __device__ __forceinline__ v8f wmma_f32(v2f a, v2f b, v8f c) {
    return __builtin_amdgcn_wmma_f32_16x16x4_f32(
        /*neg_a=*/false, a, /*neg_b=*/false, b,
        /*c_mod=*/(short)0, c, /*reuse_a=*/false, /*reuse_b=*/false);
}

// K-adjacent v2f fragment load from LDS (single ds_load_b64, even-aligned)
static __device__ __forceinline__ v2f ldsv2(const float* p) {
    return *(const v2f*)p;
}

// ---------------- kernel 0: zero GN stats ----------------
__global__ void init_stats(float* __restrict__ stats) {
    if (threadIdx.x < 16) stats[threadIdx.x] = 0.0f;
}

// ---------------- kernel 1: qkv = x @ w_qkv ----------------
// C[ROWS][QKVC] = x[ROWS][DIM] * w[DIM][QKVC]; block tile 128x64, 8 waves.
// A tile filled by async global->LDS b64 DMA; B tile stored K-adjacent.
__global__ __launch_bounds__(256) void qkv_gemm(const float* __restrict__ x,
                                                const float* __restrict__ w,
                                                float* __restrict__ qkv) {
    __shared__ float As[128][18];   // As[m][k], stride 18 (9 b64 banks, coprime 32)
    __shared__ float Bs[64][22];    // Bs[col][k], stride 22 (11 b64 banks)
    const int t = threadIdx.x;
    const int wave = t >> 5, lane = t & 31;
    const int lr = lane & 15, lh = lane >> 4;
    const int row0 = blockIdx.y * 128;
    const int col0 = blockIdx.x * 64;
    const int m0 = wave * 16;

    v8f acc[4];
#pragma unroll
    for (int j = 0; j < 4; ++j) acc[j] = vzero8();

    const int am = t >> 1;             // 0..127 : A row handled by this thread
    const int ak = (t & 1) * 8;        // 0 or 8 : first of 8 k's
    const int bk = (t >> 4) & 15;      // 0..15  : B k-row
    const int bc = (t & 15) * 4;       // 0..60  : first of 4 cols

    for (int k0 = 0; k0 < DIM; k0 += 16) {
        // ---- A tile: x[row0+am][k0+ak .. +7] -> As[am][ak..], K-contiguous
        const float* asrc = x + (size_t)(row0 + am) * DIM + k0 + ak;
#if QKV_ASYNC
        {
            const uint32_t l0 = (uint32_t)(uintptr_t)&As[am][ak];   // LDS byte addr
            const uint64_t g0 = (uint64_t)(uintptr_t)asrc;
#pragma unroll
            for (int c = 0; c < 4; ++c) {
                asm volatile("global_load_async_to_lds_b64 %0, %1, off"
                             :: "v"((uint32_t)(l0 + 8u * c)), "v"(g0 + 8ull * c)
                             : "memory");
            }
        }
#else
#pragma unroll
        for (int j = 0; j < 8; ++j) As[am][ak + j] = asrc[j];
#endif
        // ---- B tile (transposed to K-adjacent): w[k0+bk][col0+bc..+3]
        const float* bsrc = w + (size_t)(k0 + bk) * QKVC + col0 + bc;
#pragma unroll
        for (int j = 0; j < 4; ++j) Bs[bc + j][bk] = bsrc[j];

        // prefetch next k tile while this one is consumed
        if (k0 + 16 < DIM) {
            __builtin_prefetch(asrc + 16, 0, 0);
            __builtin_prefetch(bsrc + 16 * QKVC, 0, 0);
        }
#if QKV_ASYNC
        asm volatile("s_wait_asynccnt 0x0" ::: "memory");
#endif
        __syncthreads();
#pragma unroll
        for (int kk = 0; kk < 16; kk += 4) {
            const v2f a = ldsv2(&As[m0 + lr][kk + 2 * lh]);
#pragma unroll
            for (int j = 0; j < 4; ++j) {
                const v2f bb = ldsv2(&Bs[j * 16 + lr][kk + 2 * lh]);
                acc[j] = wmma_f32(a, bb, acc[j]);
            }
        }
        __syncthreads();
    }
#pragma unroll
    for (int j = 0; j < 4; ++j)
#pragma unroll
        for (int r = 0; r < 8; ++r) {
            const int row = row0 + m0 + r + 8 * lh;
            const int col = col0 + j * 16 + lr;
            qkv[(size_t)row * QKVC + col] = acc[j][r];
        }
}

// ---------------- kernel 2: softmax of k over token dim n ----------------
__global__ __launch_bounds__(256) void k_softmax(float* __restrict__ qkv) {
    const int b  = blockIdx.y;
    const int ch = blockIdx.x;
    float* base = qkv + (size_t)b * NTOK * QKVC + HID + ch;
    __shared__ float red[256];
    const int t = threadIdx.x;

    float m = -INFINITY;
    for (int n = t; n < NTOK; n += 256) m = fmaxf(m, base[(size_t)n * QKVC]);
    red[t] = m; __syncthreads();
    for (int s = 128; s > 0; s >>= 1) { if (t < s) red[t] = fmaxf(red[t], red[t + s]); __syncthreads(); }
    m = red[0]; __syncthreads();

    float sum = 0.0f;
    for (int n = t; n < NTOK; n += 256) sum += expf(base[(size_t)n * QKVC] - m);
    red[t] = sum; __syncthreads();
    for (int s = 128; s > 0; s >>= 1) { if (t < s) red[t] += red[t + s]; __syncthreads(); }
    const float inv = 1.0f / red[0];

    for (int n = t; n < NTOK; n += 256) {
        const size_t idx = (size_t)n * QKVC;
        base[idx] = expf(base[idx] - m) * inv;
    }
}

// ---------------- kernel 3: softmax of q over feature dim d, * DH^-0.5 -----
__global__ __launch_bounds__(256) void q_softmax(float* __restrict__ qkv) {
    const int t = threadIdx.x;
    const int wave = t >> 5, lane = t & 31;
    const long rowid = (long)blockIdx.x * 8 + wave;   // (b*NTOK+n)*HEADS + h
    const int  h  = (int)(rowid % HEADS);
    const long bn = rowid / HEADS;
    float* base = qkv + (size_t)bn * QKVC + h * DH;

    float v0 = base[lane], v1 = base[lane + 32];
    float m = fmaxf(v0, v1);
#pragma unroll
    for (int off = 16; off >= 1; off >>= 1) m = fmaxf(m, __shfl_xor(m, off, 32));
    const float e0 = expf(v0 - m), e1 = expf(v1 - m);
    float s = e0 + e1;
#pragma unroll
    for (int off = 16; off >= 1; off >>= 1) s += __shfl_xor(s, off, 32);
    const float sc = 0.125f / s;   // DH^-0.5 = 1/8
    base[lane]      = e0 * sc;
    base[lane + 32] = e1 * sc;
}

// ---------------- kernel 4: ctx[d][e] = sum_n k_sm[d,n] * v[e,n] ------------
// tiles stored K(=n)-adjacent so fragments are single ds_load_b64
__global__ __launch_bounds__(256) void context_gemm(const float* __restrict__ qkv,
                                                    float* __restrict__ ctx) {
    const int bh = blockIdx.x;
    const int b = bh >> 3, h = bh & 7;
    const float* kbase = qkv + (size_t)b * NTOK * QKVC + HID + h * DH;
    const float* vbase = qkv + (size_t)b * NTOK * QKVC + 2 * HID + h * DH;
    __shared__ float Kt[DH][34];   // Kt[d][n], stride 34 (17 b64 banks)
    __shared__ float Vt[DH][34];   // Vt[e][n]
    const int t = threadIdx.x;
    const int wave = t >> 5, lane = t & 31;
    const int lr = lane & 15, lh = lane >> 4;
    const int d0 = (wave & 3) * 16;
    const int e0 = (wave >> 2) * 32;

    v8f acc[2];
    acc[0] = vzero8(); acc[1] = vzero8();

    const int nn = t >> 3;          // 0..31 token within tile
    const int c  = (t & 7) * 8;     // 0..56 first channel

    for (int n0 = 0; n0 < NTOK; n0 += 32) {
        const float* ksrc = kbase + (size_t)(n0 + nn) * QKVC + c;
        const float* vsrc = vbase + (size_t)(n0 + nn) * QKVC + c;
#pragma unroll
        for (int j = 0; j < 8; ++j) { Kt[c + j][nn] = ksrc[j]; Vt[c + j][nn] = vsrc[j]; }
        __syncthreads();
#pragma unroll
        for (int kk = 0; kk < 32; kk += 4) {
            const v2f a = ldsv2(&Kt[d0 + lr][kk + 2 * lh]);        // A[d][n]
#pragma unroll
            for (int j = 0; j < 2; ++j) {
                const v2f bb = ldsv2(&Vt[e0 + j * 16 + lr][kk + 2 * lh]); // B[n][e]=v^T
                acc[j] = wmma_f32(a, bb, acc[j]);
            }
        }
        __syncthreads();
    }
    float* cdst = ctx + (size_t)bh * DH * DH;
#pragma unroll
    for (int j = 0; j < 2; ++j)
#pragma unroll
        for (int r = 0; r < 8; ++r)
            cdst[(size_t)(d0 + r + 8 * lh) * DH + e0 + j * 16 + lr] = acc[j][r];
}

// ---------------- kernel 5: out^T[n][e] = sum_d q_sm[d,n]*ctx[d][e] ---------
// writes outb[b][n][h*DH+e]; fused per-sample sum/sumsq for GroupNorm
__global__ __launch_bounds__(256) void out_gemm(const float* __restrict__ qkv,
                                                const float* __restrict__ ctx,
                                                float* __restrict__ outb,
                                                float* __restrict__ stats) {
    const int bh = blockIdx.y;
    const int b = bh >> 3, h = bh & 7;
    const int n0 = blockIdx.x * 128;
    const float* qbase = qkv + (size_t)b * NTOK * QKVC + h * DH;  // q[n][d], d contig
    __shared__ float Ct[DH][66];    // Ct[e][d] (transposed ctx), stride 66
    __shared__ float Qs[128][18];   // Qs[n][d], stride 18
    __shared__ float rsum[256], rsq[256];
    const int t = threadIdx.x;
    const int wave = t >> 5, lane = t & 31;
    const int lr = lane & 15, lh = lane >> 4;
    const int m0 = wave * 16;

    {   // stage ctx transposed: Ct[e][d] = ctx[d][e]
        const float* csrc = ctx + (size_t)bh * DH * DH;
        for (int i = t; i < DH * DH; i += 256) Ct[i & 63][i >> 6] = csrc[i];
    }
    v8f acc[4];
#pragma unroll
    for (int j = 0; j < 4; ++j) acc[j] = vzero8();

    const int qm = t >> 1;
    const int qd = (t & 1) * 8;

    for (int k0 = 0; k0 < DH; k0 += 16) {
        const float* src = qbase + (size_t)(n0 + qm) * QKVC + k0 + qd;
#pragma unroll
        for (int j = 0; j < 8; ++j) Qs[qm][qd + j] = src[j];
        __syncthreads();
#pragma unroll
        for (int kk = 0; kk < 16; kk += 4) {
            const v2f a = ldsv2(&Qs[m0 + lr][kk + 2 * lh]);        // A[n][d]=q^T
#pragma unroll
            for (int j = 0; j < 4; ++j) {
                const v2f bb = ldsv2(&Ct[j * 16 + lr][k0 + kk + 2 * lh]); // B[d][e]
                acc[j] = wmma_f32(a, bb, acc[j]);
            }
        }
        __syncthreads();
    }

    float lsum = 0.0f, lsq = 0.0f;
#pragma unroll
    for (int j = 0; j < 4; ++j)
#pragma unroll
        for (int r = 0; r < 8; ++r) {
            const int n = n0 + m0 + r + 8 * lh;
            const int e = j * 16 + lr;
            const float v = acc[j][r];
            outb[((size_t)b * NTOK + n) * HID + h * DH + e] = v;
            lsum += v; lsq += v * v;
        }
    rsum[t] = lsum; rsq[t] = lsq;
    __syncthreads();
    for (int s = 128; s > 0; s >>= 1) {
        if (t < s) { rsum[t] += rsum[t + s]; rsq[t] += rsq[t + s]; }
        __syncthreads();
    }
    if (t == 0) {
        atomicAdd(&stats[2 * b + 0], rsum[0]);
        atomicAdd(&stats[2 * b + 1], rsq[0]);
    }
}

// ---------------- kernel 6: y = GN(outb) @ w_out + b_out --------------------
// GroupNorm affine folded into A-tile load; block tile 128x64
__global__ __launch_bounds__(256) void final_gemm(const float* __restrict__ outb,
                                                  const float* __restrict__ wout,
                                                  const float* __restrict__ bout,
                                                  const float* __restrict__ gsc,
                                                  const float* __restrict__ gbi,
                                                  const float* __restrict__ stats,
                                                  float* __restrict__ y) {
    __shared__ float As[128][18];   // normalized A[m][k]
    __shared__ float Bs[64][22];    // Bs[col][k]
    const int t = threadIdx.x;
    const int wave = t >> 5, lane = t & 31;
    const int lr = lane & 15, lh = lane >> 4;
    const int row0 = blockIdx.y * 128;
    const int col0 = blockIdx.x * 64;
    const int m0 = wave * 16;

    const int b = row0 >> 12;           // 4096 rows per sample
    const float cnt = (float)NTOK * (float)HID;
    const float mu  = stats[2 * b] / cnt;
    const float var = stats[2 * b + 1] / cnt - mu * mu;
    const float rs  = rsqrtf(var + EPS_GN);

    v8f acc[4];
#pragma unroll
    for (int j = 0; j < 4; ++j) acc[j] = vzero8();

    const int am = t >> 1;
    const int ak = (t & 1) * 8;
    const int bk = (t >> 4) & 15;
    const int bc = (t & 15) * 4;

    for (int k0 = 0; k0 < HID; k0 += 16) {
        {
            const float* src = outb + (size_t)(row0 + am) * HID + k0 + ak;
#pragma unroll
            for (int j = 0; j < 8; ++j) {
                const int kc = k0 + ak + j;
                As[am][ak + j] = (src[j] - mu) * rs * gsc[kc] + gbi[kc];
            }
            if (k0 + 16 < HID) __builtin_prefetch(src + 16, 0, 0);
        }
        {
            const float* src = wout + (size_t)(k0 + bk) * DIM + col0 + bc;
#pragma unroll
            for (int j = 0; j < 4; ++j) Bs[bc + j][bk] = src[j];
            if (k0 + 16 < HID) __builtin_prefetch(src + 16 * DIM, 0, 0);
        }
        __syncthreads();
#pragma unroll
        for (int kk = 0; kk < 16; kk += 4) {
            const v2f a = ldsv2(&As[m0 + lr][kk + 2 * lh]);
#pragma unroll
            for (int j = 0; j < 4; ++j) {
                const v2f bb = ldsv2(&Bs[j * 16 + lr][kk + 2 * lh]);
                acc[j] = wmma_f32(a, bb, acc[j]);
            }
        }
        __syncthreads();
    }
#pragma unroll
    for (int j = 0; j < 4; ++j)
#pragma unroll
        for (int r = 0; r < 8; ++r) {
            const int row = row0 + m0 + r + 8 * lh;
            const int col = col0 + j * 16 + lr;
            y[(size_t)row * DIM + col] = acc[j][r] + bout[col];
        }
}

// ---------------- host-side launcher ----------------
extern "C" void kernel_launch(void* const* d_in, const int* in_sizes, int n_in,
                              void* d_out, int out_size, void* d_ws, size_t ws_size,
                              hipStream_t stream) {
    (void)in_sizes; (void)n_in; (void)out_size; (void)ws_size;
    const float* x        = (const float*)d_in[0];
    const float* w_qkv    = (const float*)d_in[1];
    const float* gn_scale = (const float*)d_in[2];
    const float* gn_bias  = (const float*)d_in[3];
    const float* w_out    = (const float*)d_in[4];
    const float* b_out    = (const float*)d_in[5];
    float* y = (float*)d_out;

    // workspace layout (floats): qkv | outb | ctx | stats  (~270 MB total)
    float* qkv   = (float*)d_ws;
    float* outb  = qkv  + (size_t)ROWS * QKVC;
    float* ctx   = outb + (size_t)ROWS * HID;
    float* stats = ctx  + (size_t)B_ * HEADS * DH * DH;

    init_stats<<<1, 32, 0, stream>>>(stats);
    qkv_gemm<<<dim3(QKVC / 64, ROWS / 128), 256, 0, stream>>>(x, w_qkv, qkv);
    k_softmax<<<dim3(HID, B_), 256, 0, stream>>>(qkv);
    q_softmax<<<(ROWS * HEADS) / 8, 256, 0, stream>>>(qkv);
    context_gemm<<<B_ * HEADS, 256, 0, stream>>>(qkv, ctx);
    out_gemm<<<dim3(NTOK / 128, B_ * HEADS), 256, 0, stream>>>(qkv, ctx, outb, stats);
    final_gemm<<<dim3(DIM / 64, ROWS / 128), 256, 0, stream>>>(outb, w_out, b_out,
                                                               gn_scale, gn_bias, stats, y);
}